// TensorGNAN_44873818308786
// MI455X (gfx1250) — compile-verified
//
#include <hip/hip_runtime.h>

// Problem constants (match reference).
constexpr int NN = 2048;   // nodes
constexpr int FD = 64;     // features
constexpr int HD = 64;     // hidden
constexpr int CD = 8;      // classes

constexpr int JSLICES = 8;             // j-split for kernel B
constexpr int JSLICE  = NN / JSLICES;  // 256 j per block
constexpr int WJ      = 64;            // j per wave
constexpr int MSTRIDE = 65;            // LDS row stride (bank-conflict pad)

typedef float v2f __attribute__((ext_vector_type(2)));
typedef float v8f __attribute__((ext_vector_type(8)));

// ---------------------------------------------------------------------------
// Kernel A: g[n,c] = sum_f ( sum_h relu(x[n,f]*W1[f,h]+b1[f,h]) * W2[f,h,c] )
//                  + sum_f b2[f,c]
// ---------------------------------------------------------------------------
__global__ __launch_bounds__(128) void gnan_feature_kernel(
    const float* __restrict__ x,  const float* __restrict__ W1,
    const float* __restrict__ b1, const float* __restrict__ W2,
    const float* __restrict__ b2, float* __restrict__ g)
{
    __shared__ float W1s[FD * HD];
    __shared__ float b1s[FD * HD];
    const int tid = threadIdx.x;
    for (int idx = tid; idx < FD * HD; idx += 128) {
        W1s[idx] = W1[idx];
        b1s[idx] = b1[idx];
    }
    __syncthreads();

    const int n = blockIdx.x * 128 + tid;
    float acc[CD];
#pragma unroll
    for (int c = 0; c < CD; ++c) acc[c] = 0.0f;

    for (int f = 0; f < FD; ++f) {
        const float xv = x[n * FD + f];
        const float* w1p = &W1s[f * HD];
        const float* b1p = &b1s[f * HD];
        const float4* w2p = (const float4*)(W2 + (size_t)f * HD * CD);
#pragma unroll 8
        for (int h = 0; h < HD; ++h) {
            float t = fmaf(xv, w1p[h], b1p[h]);
            t = fmaxf(t, 0.0f);
            const float4 wa = w2p[h * 2 + 0];
            const float4 wb = w2p[h * 2 + 1];
            acc[0] = fmaf(t, wa.x, acc[0]);
            acc[1] = fmaf(t, wa.y, acc[1]);
            acc[2] = fmaf(t, wa.z, acc[2]);
            acc[3] = fmaf(t, wa.w, acc[3]);
            acc[4] = fmaf(t, wb.x, acc[4]);
            acc[5] = fmaf(t, wb.y, acc[5]);
            acc[6] = fmaf(t, wb.z, acc[6]);
            acc[7] = fmaf(t, wb.w, acc[7]);
        }
    }
#pragma unroll 8
    for (int f = 0; f < FD; ++f) {
#pragma unroll
        for (int c = 0; c < CD; ++c) acc[c] += b2[f * CD + c];
    }
#pragma unroll
    for (int c = 0; c < CD; ++c) g[n * CD + c] = acc[c];
}

// ---------------------------------------------------------------------------
// Kernel B: fused  m[i,j] = MLP(nd[i,j]/nm[i,j])  and  partial out = m @ g
// Block = 4 waves; wave w handles rows [i0,i0+16) x cols [j0b+w*64, +64).
// m tile is computed in registers, staged in padded LDS, consumed by
// v_wmma_f32_16x16x4_f32 against a zero-padded g tile (cols 8..15 = 0).
// Partial results per j-slice -> workspace (deterministic, no atomics).
// ---------------------------------------------------------------------------
__global__ __launch_bounds__(128) void gnan_msg_wmma_kernel(
    const float* __restrict__ nd,  const float* __restrict__ nm,
    const float* __restrict__ Wm1, const float* __restrict__ bm1,
    const float* __restrict__ Wm2, const float* __restrict__ bm2,
    const float* __restrict__ g,   float* __restrict__ part)
{
    __shared__ float mlds[4 * 16 * MSTRIDE];  // per-wave 16x64 m tiles, padded
    __shared__ float gpad[JSLICE * 16];       // g slice padded to 16 columns
    __shared__ float ctile[4 * 256];          // per-wave 16x16 C tiles

    const int tid  = threadIdx.x;
    const int wave = tid >> 5;
    const int lane = tid & 31;
    const int i0   = blockIdx.x * 16;
    const int j0b  = blockIdx.y * JSLICE;

    // Stage g[j0b .. j0b+255, 0..7] zero-padded to 16 cols (exact trip count:
    // 4096/128 = 32 iterations per thread, no divergence).
    for (int idx = tid; idx < JSLICE * 16; idx += 128) {
        const int jj = idx >> 4;
        const int c  = idx & 15;
        gpad[idx] = (c < CD) ? g[(size_t)(j0b + jj) * CD + c] : 0.0f;
    }

    // ---- per-wave m tile: 16 rows x 64 cols, 32 elements per lane ----
    const int j0 = j0b + wave * WJ;
    float dv[32], ac[32];
#pragma unroll
    for (int e = 0; e < 32; ++e) {
        const int idx = e * 32 + lane;
        const int r   = idx >> 6;
        const int col = idx & 63;
        const size_t gi = (size_t)(i0 + r) * NN + (size_t)(j0 + col);
        const float nd_v = __builtin_nontemporal_load(&nd[gi]);
        const float nm_v = __builtin_nontemporal_load(&nm[gi]);
        dv[e] = nd_v / nm_v;
        ac[e] = 0.0f;
    }
    const float bm2v = bm2[0];
#pragma unroll 4
    for (int h = 0; h < HD; ++h) {
        const float w1h = Wm1[h];   // wave-uniform -> scalar loads
        const float bbh = bm1[h];
        const float w2h = Wm2[h];
#pragma unroll
        for (int e = 0; e < 32; ++e) {
            float t = fmaf(dv[e], w1h, bbh);
            t = fmaxf(t, 0.0f);
            ac[e] = fmaf(t, w2h, ac[e]);
        }
    }
    float* mw = &mlds[wave * 16 * MSTRIDE];
#pragma unroll
    for (int e = 0; e < 32; ++e) {
        const int idx = e * 32 + lane;
        const int r   = idx >> 6;
        const int col = idx & 63;
        mw[r * MSTRIDE + col] = ac[e] + bm2v;
    }
    __syncthreads();  // gpad visible to all; mlds is per-wave (in-order DS)

    // ---- WMMA K-loop: 16 steps of V_WMMA_F32_16X16X4_F32 ----
    // A (16x4 f32): lanes 0-15 rows M=0..15 with K=k0,k0+1; lanes 16-31 K=k0+2,k0+3.
    // B (4x16 f32): lanes 0-15 cols N=0..15 with K=k0,k0+1; lanes 16-31 K=k0+2,k0+3.
    //
    // Hoist the lane-half select ONCE so the 16-step loop indexes two fixed
    // base pointers with affine offsets -> DS immediate offsets, no per-step
    // v_cndmask/v_or address recomputation in the hot loop.
    v8f cacc = {0.f, 0.f, 0.f, 0.f, 0.f, 0.f, 0.f, 0.f};
    const int rrow = (lane < 16) ? lane : (lane - 16);
    const int ncol = lane & 15;
    const int koff = (lane < 16) ? 0 : 2;              // K pair owned by this half
    const float* __restrict__ aptr = mw + rrow * MSTRIDE + koff;
    const float* __restrict__ bptr = gpad + (wave * WJ + koff) * 16 + ncol;
#pragma unroll
    for (int ks = 0; ks < WJ / 4; ++ks) {
        v2f a, b;
        a.x = aptr[ks * 4 + 0];        // K = 4*ks + koff
        a.y = aptr[ks * 4 + 1];        // K = 4*ks + koff + 1
        b.x = bptr[ks * 64 + 0];       // row (4*ks+koff)   of gpad
        b.y = bptr[ks * 64 + 16];      // row (4*ks+koff+1) of gpad
        cacc = __builtin_amdgcn_wmma_f32_16x16x4_f32(
            false, a, false, b, (short)0, cacc, false, false);
    }

    // C layout: VGPR r, lane l -> row = (l<16 ? r : r+8), col = l%16.
#pragma unroll
    for (int r = 0; r < 8; ++r) {
        const int row = (lane < 16) ? r : (r + 8);
        ctile[wave * 256 + row * 16 + ncol] = cacc[r];
    }
    __syncthreads();

    // Combine 4 waves' tiles, keep valid cols 0..7, write j-slice partial.
    for (int idx = tid; idx < 256; idx += 128) {
        const int row = idx >> 4;
        const int col = idx & 15;
        if (col < CD) {
            const float s = ctile[idx] + ctile[256 + idx] +
                            ctile[512 + idx] + ctile[768 + idx];
            part[(size_t)blockIdx.y * (NN * CD) + (size_t)(i0 + row) * CD + col] = s;
        }
    }
}

// ---------------------------------------------------------------------------
// Kernel C: out = sum over j-slice partials (deterministic reduction).
// ---------------------------------------------------------------------------
__global__ __launch_bounds__(256) void gnan_reduce_kernel(
    const float* __restrict__ part, float* __restrict__ out)
{
    const int t = blockIdx.x * 256 + threadIdx.x;  // 0 .. NN*CD-1
    float s = 0.0f;
#pragma unroll
    for (int p = 0; p < JSLICES; ++p) s += part[(size_t)p * (NN * CD) + t];
    out[t] = s;
}

// ---------------------------------------------------------------------------
extern "C" void kernel_launch(void* const* d_in, const int* in_sizes, int n_in,
                              void* d_out, int out_size, void* d_ws, size_t ws_size,
                              hipStream_t stream) {
    (void)in_sizes; (void)n_in; (void)out_size; (void)ws_size;
    const float* x   = (const float*)d_in[0];
    const float* nd  = (const float*)d_in[1];
    const float* nm  = (const float*)d_in[2];
    const float* W1  = (const float*)d_in[3];
    const float* b1  = (const float*)d_in[4];
    const float* W2  = (const float*)d_in[5];
    const float* b2  = (const float*)d_in[6];
    const float* Wm1 = (const float*)d_in[7];
    const float* bm1 = (const float*)d_in[8];
    const float* Wm2 = (const float*)d_in[9];
    const float* bm2 = (const float*)d_in[10];
    float* out = (float*)d_out;

    float* g    = (float*)d_ws;          // NN*CD floats         (64 KB)
    float* part = g + (size_t)NN * CD;   // JSLICES*NN*CD floats (512 KB)

    gnan_feature_kernel<<<dim3(NN / 128), dim3(128), 0, stream>>>(
        x, W1, b1, W2, b2, g);
    gnan_msg_wmma_kernel<<<dim3(NN / 16, JSLICES), dim3(128), 0, stream>>>(
        nd, nm, Wm1, bm1, Wm2, bm2, g, part);
    gnan_reduce_kernel<<<dim3((NN * CD) / 256), dim3(256), 0, stream>>>(
        part, out);
}